// WealthRNN_14894946583097
// MI455X (gfx1250) — compile-verified
//
#include <hip/hip_runtime.h>
#include <cstdint>

#ifndef __has_builtin
#define __has_builtin(x) 0
#endif

// ---- CDNA5 async-copy availability probes (compile-time) -------------------
#if __has_builtin(__builtin_amdgcn_global_load_async_to_lds_b128)
#define USE_ASYNC 1
#else
#define USE_ASYNC 0
#endif

#if __has_builtin(__builtin_amdgcn_s_wait_asynccnt)
#define WAIT_ASYNCCNT(n) __builtin_amdgcn_s_wait_asynccnt(n)
#else
#define WAIT_ASYNCCNT(n) asm volatile("s_wait_asynccnt %0" ::"i"(n) : "memory")
#endif

#define THREADS 64    // 2 wave32 per block
#define ELEMS   4     // batch lanes per thread (b128 path)
#define DEPTH   32    // async prefetch depth in steps (2/step -> ~62 outstanding, ASYNCcnt max)
#define PFDIST  96    // global_prefetch distance (steps): ~1.5us of compute ahead -> HBM->L2 staging

// clang vector type (works with __builtin_nontemporal_store, supports .x/.y/.z/.w)
typedef float v4f __attribute__((ext_vector_type(4)));

// 16-byte int vector matching the async builtin's pointee type
typedef int v4i __attribute__((vector_size(16)));
typedef __attribute__((address_space(1))) v4i gv4i;   // global (__device__)
typedef __attribute__((address_space(3))) v4i lv4i;   // LDS (__shared__)

__device__ __forceinline__ void async_cp16(const float* gsrc, float* ldst) {
#if USE_ASYNC
  // GLOBAL_LOAD_ASYNC_TO_LDS_B128: per-lane 16B global -> LDS, tracked by ASYNCcnt
  __builtin_amdgcn_global_load_async_to_lds_b128((gv4i*)gsrc, (lv4i*)ldst, 0, 0);
#else
  (void)gsrc; (void)ldst;
#endif
}

__device__ __forceinline__ float rcp_fast(float x) {
#if __has_builtin(__builtin_amdgcn_rcpf)
  return __builtin_amdgcn_rcpf(x);   // raw v_rcp_f32, shortens the serial chain
#else
  return 1.0f / x;
#endif
}

// One step of the no-trade-region RNN for one batch lane.
// c0 = b_h - pi_bar ; c2 = 2*b_h ; c3 = pi_bar + b_h
__device__ __forceinline__ float rnn_step(float h, float x, float r,
                                          float wIn, float wH, float wF1, float wF2,
                                          float c0, float c2, float c3) {
  float a    = 1.0f + r;                       // off-chain (depends only on r)
  float den  = __builtin_fmaf(h, r, 1.0f);     // chain
  float hadj = (h * a) * rcp_fast(den);        // chain (rcp is TRANS)
  float p    = __builtin_fmaf(x, wIn, c0);     // off-chain
  float ing  = __builtin_fmaf(wH, hadj, p);    // chain
  float i2   = __builtin_fmaf(wF1, fmaxf(ing, 0.0f), c2);
  return       __builtin_fmaf(wF2, fmaxf(i2,  0.0f), c3);
}

__global__ __launch_bounds__(THREADS)
void wealth_rnn_kernel(const float* __restrict__ inputs,   // [S,B]
                       const float* __restrict__ returns,  // [S-1,B]
                       const float* __restrict__ target,
                       const float* __restrict__ w_in,  const float* __restrict__ w_h,
                       const float* __restrict__ b_h,   const float* __restrict__ w_fc1,
                       const float* __restrict__ w_fc2,
                       float* __restrict__ out,            // [S,B] then [B] (hT)
                       int S, int Btot) {
  const int tid  = blockIdx.x * blockDim.x + threadIdx.x;
  const int base = tid * ELEMS;
  if (base >= Btot) return;

  // Uniform scalar parameters (hoisted out of the loop by the compiler).
  const float pi_bar = target[0];
  const float wIn = w_in[0], wH = w_h[0], bH = b_h[0];
  const float wF1 = w_fc1[0], wF2 = w_fc2[0];
  const float c0 = bH - pi_bar;
  const float c2 = 2.0f * bH;
  const float c3 = pi_bar + bH;

  // h0 = inputs[0]; it is also output[0].
  v4f h4 = *(const v4f*)(inputs + base);
  __builtin_nontemporal_store(h4, (v4f*)(out + base));
  float h_0 = h4.x, h_1 = h4.y, h_2 = h4.z, h_3 = h4.w;

  auto step4 = [&](const v4f& x4, const v4f& r4) {
    h_0 = rnn_step(h_0, x4.x, r4.x, wIn, wH, wF1, wF2, c0, c2, c3);
    h_1 = rnn_step(h_1, x4.y, r4.y, wIn, wH, wF1, wF2, c0, c2, c3);
    h_2 = rnn_step(h_2, x4.z, r4.z, wIn, wH, wF1, wF2, c0, c2, c3);
    h_3 = rnn_step(h_3, x4.w, r4.w, wIn, wH, wF1, wF2, c0, c2, c3);
  };

  // Untracked HBM->L2 staging far ahead of the tracked async pipeline.
  auto prefetch_far = [&](int t) {
    if (t + PFDIST < S) {
      __builtin_prefetch(inputs  + (size_t)(t + PFDIST) * Btot + base, 0, 1);
      __builtin_prefetch(returns + (size_t)(t + PFDIST - 1) * Btot + base, 0, 1);
    }
  };

#if USE_ASYNC
  // Per-wave LDS ring: DEPTH slots, each slot = 32 lanes * (4 x-floats + 4 r-floats).
  // Each wave reads only data it issued -> ASYNCcnt wait suffices, no barriers.
  __shared__ float lds[(THREADS / 32) * DEPTH * 256];
  const int wave  = threadIdx.x >> 5;
  const int lane  = threadIdx.x & 31;
  const int wbase = wave * (DEPTH * 256);

  auto issue = [&](int t, int slot) {
    async_cp16(inputs  + (size_t)t       * Btot + base, &lds[wbase + slot * 256 +       lane * 4]);
    async_cp16(returns + (size_t)(t - 1) * Btot + base, &lds[wbase + slot * 256 + 128 + lane * 4]);
  };
  auto lds_read = [&](int slot, v4f& x4, v4f& r4) {
    x4 = *(const v4f*)&lds[wbase + slot * 256 +       lane * 4];
    r4 = *(const v4f*)&lds[wbase + slot * 256 + 128 + lane * 4];
  };

  // Prologue: fill the pipeline (steps 1..DEPTH).  Requires S > DEPTH+1 (S=1024).
#pragma unroll
  for (int t = 1; t <= DEPTH; ++t) issue(t, (t - 1) & (DEPTH - 1));

  // Prime the register stage with step 1 (oldest pair: ASYNCcnt <= 2*(DEPTH-1)).
  v4f xc, rc, xn, rn;
  WAIT_ASYNCCNT(2 * (DEPTH - 1));
  lds_read(0, xc, rc);

  int t = 1;
  // Steady state. Invariant at top of iter t: regs hold step t; issued through
  // step t-1+DEPTH. Step t+1 is the 2nd-oldest outstanding pair -> done when
  // ASYNCcnt <= 2*(DEPTH-2). DS latency of the t+1 read hides under compute(t).
  for (; t < S - DEPTH; ++t) {
    WAIT_ASYNCCNT(2 * (DEPTH - 2));
    lds_read(t & (DEPTH - 1), xn, rn);              // step t+1
    step4(xc, rc);                                  // step t
    v4f o = {h_0, h_1, h_2, h_3};
    __builtin_nontemporal_store(o, (v4f*)(out + (size_t)t * Btot + base));
    prefetch_far(t);
    issue(t + DEPTH, (t - 1) & (DEPTH - 1));        // reuse slot just retired
    xc = xn; rc = rn;
  }
  // Tail: pipeline drains (no new issues); wait for everything outstanding.
  for (; t < S - 1; ++t) {
    WAIT_ASYNCCNT(0);
    lds_read(t & (DEPTH - 1), xn, rn);              // step t+1
    step4(xc, rc);
    v4f o = {h_0, h_1, h_2, h_3};
    __builtin_nontemporal_store(o, (v4f*)(out + (size_t)t * Btot + base));
    xc = xn; rc = rn;
  }
  // Final step t = S-1.
  step4(xc, rc);
  {
    v4f o = {h_0, h_1, h_2, h_3};
    __builtin_nontemporal_store(o, (v4f*)(out + (size_t)(S - 1) * Btot + base));
  }
#else
  // Fallback: 1-step register pipeline + global_prefetch_b8 far ahead.
  v4f xn = *(const v4f*)(inputs + (size_t)Btot + base);
  v4f rn = *(const v4f*)(returns + base);
  for (int t = 1; t < S; ++t) {
    v4f x4 = xn, r4 = rn;
    if (t + 1 < S) {
      xn = *(const v4f*)(inputs + (size_t)(t + 1) * Btot + base);
      rn = *(const v4f*)(returns + (size_t)t * Btot + base);
    }
    prefetch_far(t);
    step4(x4, r4);
    v4f o = {h_0, h_1, h_2, h_3};
    __builtin_nontemporal_store(o, (v4f*)(out + (size_t)t * Btot + base));
  }
#endif

  // Second tuple element: hT (= state after last step) at offset S*B.
  v4f oT = {h_0, h_1, h_2, h_3};
  *(v4f*)(out + (size_t)S * Btot + base) = oT;
}

extern "C" void kernel_launch(void* const* d_in, const int* in_sizes, int n_in,
                              void* d_out, int out_size, void* d_ws, size_t ws_size,
                              hipStream_t stream) {
  (void)n_in; (void)out_size; (void)d_ws; (void)ws_size;
  const float* inputs  = (const float*)d_in[0];
  const float* returns = (const float*)d_in[1];
  const float* target  = (const float*)d_in[2];
  const float* w_in    = (const float*)d_in[3];
  const float* w_h     = (const float*)d_in[4];
  const float* b_h     = (const float*)d_in[5];
  const float* w_fc1   = (const float*)d_in[6];
  const float* w_fc2   = (const float*)d_in[7];

  // inputs = S*B, returns = (S-1)*B  =>  B = diff, S = inputs/B
  const int B = in_sizes[0] - in_sizes[1];   // 16384
  const int S = in_sizes[0] / B;             // 1024

  const int nthreads = B / ELEMS;            // 4096 threads = 128 wave32
  const int blocks   = (nthreads + THREADS - 1) / THREADS;  // 64 blocks

  wealth_rnn_kernel<<<blocks, THREADS, 0, stream>>>(
      inputs, returns, target, w_in, w_h, b_h, w_fc1, w_fc2,
      (float*)d_out, S, B);
}